// DL_R_sum_MRC_62113817035446
// MI455X (gfx1250) — compile-verified
//
#include <hip/hip_runtime.h>

// ---------------------------------------------------------------------------
// DL_R_sum_MRC on MI455X (gfx1250, wave32).
// Memory-bound (~447 MB streamed, ~0.5 GFLOP -> HBM floor ~19us @ 23.3 TB/s).
// Per (b,RB) tile: D(16x16) = A(16x64) * B(64x16) via 16 x V_WMMA_F32_16X16X4_F32
// (f32 in/out). Operands are staged in LDS in
// fragment-native layout so the inner loop is branch-free:
//   per k-step: ds_load_b64 (A pair) + ds_load_b64 (B pair) + v_wmma.
// ---------------------------------------------------------------------------

typedef __attribute__((ext_vector_type(2))) float v2f;
typedef __attribute__((ext_vector_type(8))) float v8f;

#define N_RB    13
#define B_TOT   4096
#define TILES   (B_TOT * N_RB)
#define APAD    66   // row stride (words): 8B-aligned rows, breaks bank striding

// One wave32 per (b, RB-i) tile.
__global__ __launch_bounds__(32) void mrc_rate_tile_kernel(
    const float* __restrict__ H1,   // [B,52,32,2,2]
    const float* __restrict__ H2,   // [B,52,32,2,2]
    const float* __restrict__ P,    // [B,13,2]
    const float* __restrict__ yp,   // [B,64,13,2]
    float* __restrict__ partial)    // [TILES]
{
    // A[M=(j,k,r)][kk = comp*32 + t]  (16 x 64, padded)
    __shared__ float sA[16][APAD];
    // B^T[n][kk]: n0=re(c0), n1=re(c1), n2=im(c0), n3=im(c1); n>=4 zero.
    __shared__ float sBt[16][APAD];
    __shared__ float sD[16][16];    // WMMA D spill: [M][N]
    __shared__ float sR[8];         // per-(j,k) rates

    const int tile = blockIdx.x;
    const int b    = tile / N_RB;
    const int i    = tile - b * N_RB;
    const int lane = threadIdx.x;   // 0..31

    // ---- Stage A directly from coalesced global float4 loads.
    // H flat per b: sc*128 + t*4 + r*2 + comp; subcarriers i*4..i*4+3 are
    // 512 contiguous floats per user. Lane's float4 at off = j*128 + lane*4
    // is exactly H[b, i*4+j, t=lane, r=0..1, comp=0..1].
    const size_t hbase = (size_t)(b * 52 + i * 4) * 128;
    #pragma unroll
    for (int j = 0; j < 4; ++j) {
        const int off = j * 128 + lane * 4;
        const float4 x1 = *reinterpret_cast<const float4*>(H1 + hbase + off);
        const float4 x2 = *reinterpret_cast<const float4*>(H2 + hbase + off);
        sA[j * 4 + 0][lane]      = x1.x;   // k=0, r=0, re
        sA[j * 4 + 0][32 + lane] = x1.y;   // k=0, r=0, im
        sA[j * 4 + 1][lane]      = x1.z;   // k=0, r=1, re
        sA[j * 4 + 1][32 + lane] = x1.w;   // k=0, r=1, im
        sA[j * 4 + 2][lane]      = x2.x;   // k=1, r=0, re
        sA[j * 4 + 2][32 + lane] = x2.y;
        sA[j * 4 + 3][lane]      = x2.z;   // k=1, r=1, re
        sA[j * 4 + 3][32 + lane] = x2.w;
    }

    // ---- Load V(:,i,c), normalize over antenna axis (wave reduction).
    // y_pred flat: ((b*64 + c*32 + t)*13 + i)*2 + comp ; lane <-> t.
    float vr[2], vi[2];
    #pragma unroll
    for (int c = 0; c < 2; ++c) {
        const size_t vidx = ((size_t)(b * 64 + c * 32 + lane) * N_RB + i) * 2;
        vr[c] = yp[vidx];
        vi[c] = yp[vidx + 1];
    }
    float ss0 = vr[0] * vr[0] + vi[0] * vi[0];
    float ss1 = vr[1] * vr[1] + vi[1] * vi[1];
    #pragma unroll
    for (int m = 16; m > 0; m >>= 1) {
        ss0 += __shfl_xor(ss0, m, 32);
        ss1 += __shfl_xor(ss1, m, 32);
    }
    const float inv0 = rsqrtf(ss0), inv1 = rsqrtf(ss1);
    const float vr0n = vr[0] * inv0, vi0n = vi[0] * inv0;
    const float vr1n = vr[1] * inv1, vi1n = vi[1] * inv1;

    // ---- Stage B^T (signs pre-applied; unconditional stores, no branches).
    // Rows kk<32: [Vr | Vr | Vi | Vi]; rows kk>=32: [-Vi | -Vi | Vr | Vr].
    sBt[0][lane]      =  vr0n;  sBt[1][lane]      =  vr1n;
    sBt[2][lane]      =  vi0n;  sBt[3][lane]      =  vi1n;
    sBt[0][32 + lane] = -vi0n;  sBt[1][32 + lane] = -vi1n;
    sBt[2][32 + lane] =  vr0n;  sBt[3][32 + lane] =  vr1n;
    // Zero-fill dead columns n = 4..15 (once).
    #pragma unroll
    for (int q = 0; q < 24; ++q) {
        const int idx = q * 32 + lane;        // 0..767 over 12 rows x 64 cols
        sBt[4 + (idx >> 6)][idx & 63] = 0.0f;
    }

    __syncthreads();

    // ---- 16 x WMMA f32 16x16x4. ISA 32-bit operand layout:
    //   lane<16  : VGPR0=K(4s+0), VGPR1=K(4s+1), row/col = lane
    //   lane>=16 : VGPR0=K(4s+2), VGPR1=K(4s+3), row/col = lane-16
    const int M    = lane & 15;
    const int half = lane >> 4;
    const float* aRow = &sA[M][half * 2];
    const float* bRow = &sBt[M][half * 2];

    v8f acc = {};
    #pragma unroll
    for (int s = 0; s < 16; ++s) {
        const v2f afrag = *reinterpret_cast<const v2f*>(aRow + s * 4);
        const v2f bfrag = *reinterpret_cast<const v2f*>(bRow + s * 4);
        acc = __builtin_amdgcn_wmma_f32_16x16x4_f32(
            false, afrag, false, bfrag, (short)0, acc, false, false);
    }

    // ---- Spill D: VGPR v holds M = v + 8*half, N = lane&15.
    #pragma unroll
    for (int v = 0; v < 8; ++v)
        sD[v + 8 * half][M] = acc[v];

    __syncthreads();

    // ---- Epilogue: 8 lanes, one (j,k) each.
    if (lane < 8) {
        const int jj = lane >> 1;
        const int kk = lane & 1;
        const int Mb = jj * 4 + kk * 2;

        float hvre[2][2], hvim[2][2];
        #pragma unroll
        for (int r = 0; r < 2; ++r)
            #pragma unroll
            for (int c = 0; c < 2; ++c) {
                hvre[r][c] = sD[Mb + r][c];
                hvim[r][c] = sD[Mb + r][2 + c];
            }

        // HF = HV(:, c=k); WHV_c = sum_r conj(HF_r)*HV_rc / ||HF||
        const float n2 = hvre[0][kk] * hvre[0][kk] + hvim[0][kk] * hvim[0][kk]
                       + hvre[1][kk] * hvre[1][kk] + hvim[1][kk] * hvim[1][kk];
        const float invn = rsqrtf(n2);

        float pw[2];
        pw[0] = P[((size_t)b * N_RB + i) * 2 + 0];
        pw[1] = P[((size_t)b * N_RB + i) * 2 + 1];

        float sPow[2];
        #pragma unroll
        for (int c = 0; c < 2; ++c) {
            float wre = 0.0f, wim = 0.0f;
            #pragma unroll
            for (int r = 0; r < 2; ++r) {
                wre += hvre[r][kk] * hvre[r][c] + hvim[r][kk] * hvim[r][c];
                wim += hvre[r][kk] * hvim[r][c] - hvim[r][kk] * hvre[r][c];
            }
            wre *= invn;  wim *= invn;
            sPow[c] = (wre * wre + wim * wim) * pw[c];
        }
        const float nom = sPow[kk];
        const float den = sPow[0] + sPow[1] + 0.1f - nom;
        sR[lane] = log1pf(nom / den) * 1.4426950408889634f;  // log2(1+x)
    }
    __syncthreads();

    if (lane == 0) {
        float t8 = 0.0f;
        #pragma unroll
        for (int q = 0; q < 8; ++q) t8 += sR[q];
        partial[tile] = t8;
    }
}

// Deterministic final reduction: single block, fixed traversal order.
__global__ __launch_bounds__(256) void mrc_reduce_kernel(
    const float* __restrict__ in, int n, float scale, float* __restrict__ out)
{
    __shared__ float sb[256];
    float s = 0.0f;
    for (int idx = threadIdx.x; idx < n; idx += 256) s += in[idx];
    sb[threadIdx.x] = s;
    __syncthreads();
    #pragma unroll
    for (int w = 128; w > 0; w >>= 1) {
        if (threadIdx.x < (unsigned)w) sb[threadIdx.x] += sb[threadIdx.x + w];
        __syncthreads();
    }
    if (threadIdx.x == 0) out[0] = -sb[0] * scale;
}

extern "C" void kernel_launch(void* const* d_in, const int* in_sizes, int n_in,
                              void* d_out, int out_size, void* d_ws, size_t ws_size,
                              hipStream_t stream) {
    const float* H1 = (const float*)d_in[0];
    const float* H2 = (const float*)d_in[1];
    const float* P  = (const float*)d_in[2];
    const float* yp = (const float*)d_in[3];
    float* out = (float*)d_out;
    float* ws  = (float*)d_ws;   // TILES floats of scratch (208 KB)

    mrc_rate_tile_kernel<<<TILES, 32, 0, stream>>>(H1, H2, P, yp, ws);
    mrc_reduce_kernel<<<1, 256, 0, stream>>>(ws, TILES,
                                             1.0f / (float)(B_TOT * 52), out);
}